// GaborRenderer_76287209112278
// MI455X (gfx1250) — compile-verified
//
#include <hip/hip_runtime.h>
#include <cstdint>

#define SRF        24000.0f
#define MAXW       4096
#define BIN        256
#define BIN_SHIFT  8
#define CAP        1024
#define TWOPI      6.28318530717958647692f

// ---------------------------------------------------------------------------
// Window computation — mirror the reference's fp32 op sequence exactly.
// Non-contractable intrinsics so -ffp-contract cannot move a trunc() boundary.
// ---------------------------------------------------------------------------
__device__ __forceinline__ void atom_window(float tau, float sigma, int ns,
                                            int& s0, int& e) {
  float ts = __fmul_rn(tau, SRF);                    // tau_samples
  float ss = __fmul_rn(sigma, SRF);                  // sigma_samples
  float hw = fminf(fmaxf(__fmul_rn(ss, 3.0f), 1.0f), 2048.0f);
  int a = (int)__fsub_rn(ts, hw);                    // trunc toward zero
  if (a < 0) a = 0;                                  // clip start >= 0
  int b = (int)__fadd_rn(ts, hw);
  int lim = ns - 1;
  if (b > lim) b = lim;                              // end = min(.., ns-1)
  int cap = a + (MAXW - 1);                          // local index < MAX_CHUNK
  if (b > cap) b = cap;
  s0 = a;
  e  = b;
}

// CDNA5 async global->LDS gather of one f32 (per-lane addresses, GV mode).
__device__ __forceinline__ void async_ld_f32(float* lds_dst, const float* gsrc) {
  uint32_t l = (uint32_t)(uintptr_t)lds_dst;         // low 32 bits = LDS offset
  uint64_t g = (uint64_t)(uintptr_t)gsrc;
  asm volatile("global_load_async_to_lds_b32 %0, %1, off"
               :: "v"(l), "v"(g) : "memory");
}

__device__ __forceinline__ void wait_async_zero() {
  asm volatile("s_wait_asynccnt 0x0" ::: "memory");
}

// ---------------------------------------------------------------------------
// Kernel 0: zero bin counters (workspace is poisoned by harness).
// ---------------------------------------------------------------------------
__global__ void gabor_zero_i32(int* __restrict__ p, int n) {
  int i = blockIdx.x * blockDim.x + threadIdx.x;
  if (i < n) p[i] = 0;
}

__global__ void gabor_zero_f32(float* __restrict__ p, int n) {
  int i = blockIdx.x * blockDim.x + threadIdx.x;
  if (i < n) p[i] = 0.0f;
}

// ---------------------------------------------------------------------------
// Kernel 1: bin atoms by the 256-sample output tiles their window touches.
// ---------------------------------------------------------------------------
__global__ void gabor_bin_atoms(const float* __restrict__ tau,
                                const float* __restrict__ sigma,
                                int n_atoms, int ns, int nbins,
                                int* __restrict__ counts,
                                int* __restrict__ lists) {
  int i = blockIdx.x * blockDim.x + threadIdx.x;
  if (i >= n_atoms) return;
  int s0, e;
  atom_window(tau[i], sigma[i], ns, s0, e);
  int b0 = s0 >> BIN_SHIFT;
  int b1 = e  >> BIN_SHIFT;
  if (b1 >= nbins) b1 = nbins - 1;
  for (int b = b0; b <= b1; ++b) {
    int pos = atomicAdd(&counts[b], 1);
    if (pos < CAP) lists[(size_t)b * CAP + pos] = i;
  }
}

// ---------------------------------------------------------------------------
// Kernel 2: per-bin gather render. 256 threads = 256 consecutive samples.
// Raw atom params staged into LDS via async global->LDS gathers, then a
// once-per-atom LDS pass packs precomputed records:
//   q0 = {A, tau, omega, gamma/2}
//   q1 = {phi, 1/(2*sigma^2+1e-8), bitcast(s0), bitcast(e)}
// Consume loop: 2x ds_load_b128 (lane-uniform broadcast) + ~14 VALU/element.
// ---------------------------------------------------------------------------
__global__ __launch_bounds__(BIN) void gabor_render_binned(
    const float* __restrict__ amp, const float* __restrict__ tau,
    const float* __restrict__ om,  const float* __restrict__ sg,
    const float* __restrict__ ph,  const float* __restrict__ gm,
    const int* __restrict__ counts, const int* __restrict__ lists,
    float* __restrict__ out, int ns) {
  __shared__ float scrA[BIN], scrT[BIN], scrO[BIN], scrS[BIN], scrP[BIN], scrG[BIN];
  __shared__ float4 sQ0[BIN];
  __shared__ float4 sQ1[BIN];

  const int bin = blockIdx.x;
  const int t   = threadIdx.x;
  const int s   = (bin << BIN_SHIFT) + t;
  const float t_samp = (float)s / SRF;   // one fp32 div per thread, hoisted

  float acc = 0.0f;
  int cnt = counts[bin];
  if (cnt > CAP) cnt = CAP;
  const int* __restrict__ lst = lists + (size_t)bin * CAP;

  for (int base = 0; base < cnt; base += BIN) {
    const int m = (cnt - base < BIN) ? (cnt - base) : BIN;

    // --- stage raw params via async global->LDS gather -------------------
    if (t < m) {
      const int a = lst[base + t];
      async_ld_f32(&scrA[t], amp + a);
      async_ld_f32(&scrT[t], tau + a);
      async_ld_f32(&scrO[t], om  + a);
      async_ld_f32(&scrS[t], sg  + a);
      async_ld_f32(&scrP[t], ph  + a);
      async_ld_f32(&scrG[t], gm  + a);
    }
    wait_async_zero();
    __syncthreads();

    // --- once-per-atom precompute into packed records --------------------
    if (t < m) {
      const float A = scrA[t], T = scrT[t], O = scrO[t];
      const float S = scrS[t], P = scrP[t], G = scrG[t];
      int s0, e;
      atom_window(T, S, ns, s0, e);
      const float d    = 2.0f * S * S + 1e-8f;
      const float invd = 1.0f / d;                   // per-atom, hoists the div
      sQ0[t] = make_float4(A, T, O, 0.5f * G);
      sQ1[t] = make_float4(P, invd, __int_as_float(s0), __int_as_float(e));
    }
    __syncthreads();

    // --- consume: accumulate this thread's sample over the chunk ---------
    #pragma unroll 2
    for (int j = 0; j < m; ++j) {
      const float4 q0 = sQ0[j];
      const float4 q1 = sQ1[j];
      const int s0 = __float_as_int(q1.z);
      const int e  = __float_as_int(q1.w);
      if (s >= s0 && s <= e) {
        const float tc    = t_samp - q0.y;                    // t - tau
        const float env   = __expf(-(tc * tc) * q1.y);        // exp(-t^2/(2s^2+eps))
        const float inner = tc * fmaf(q0.w, tc, q0.z);        // w*t + 0.5*g*t^2
        const float phase = fmaf(TWOPI, inner, q1.x);         // 2pi*inner + phi
        acc = fmaf(q0.x * env, __cosf(phase), acc);
      }
    }
    __syncthreads();
  }

  if (s < ns) out[s] = acc;
}

// ---------------------------------------------------------------------------
// Fallback (no/small workspace): one block per atom, atomic scatter-add.
// ---------------------------------------------------------------------------
__global__ void gabor_render_atomic(const float* __restrict__ amp,
                                    const float* __restrict__ tau,
                                    const float* __restrict__ om,
                                    const float* __restrict__ sg,
                                    const float* __restrict__ ph,
                                    const float* __restrict__ gm,
                                    float* __restrict__ out, int ns) {
  const int a = blockIdx.x;
  const float A = amp[a], T = tau[a], O = om[a], S = sg[a], P = ph[a], G = gm[a];
  int s0, e;
  atom_window(T, S, ns, s0, e);
  const float invd  = 1.0f / (2.0f * S * S + 1e-8f);
  const float halfG = 0.5f * G;
  for (int s = s0 + (int)threadIdx.x; s <= e; s += (int)blockDim.x) {
    const float tc    = (float)s / SRF - T;
    const float env   = __expf(-(tc * tc) * invd);
    const float phase = fmaf(TWOPI, tc * fmaf(halfG, tc, O), P);
    atomicAdd(&out[s], A * env * __cosf(phase));
  }
}

// ---------------------------------------------------------------------------
extern "C" void kernel_launch(void* const* d_in, const int* in_sizes, int n_in,
                              void* d_out, int out_size, void* d_ws, size_t ws_size,
                              hipStream_t stream) {
  const float* amp = (const float*)d_in[0];
  const float* tau = (const float*)d_in[1];
  const float* om  = (const float*)d_in[2];
  const float* sg  = (const float*)d_in[3];
  const float* ph  = (const float*)d_in[4];
  const float* gm  = (const float*)d_in[5];
  const int n_atoms = in_sizes[0];
  const int ns      = out_size;           // == num_samples
  float* out = (float*)d_out;

  const int nbins = (ns + BIN - 1) / BIN;
  const size_t need = (size_t)nbins * sizeof(int) * (size_t)(1 + CAP);

  if (ws_size >= need) {
    int* counts = (int*)d_ws;
    int* lists  = counts + nbins;
    gabor_zero_i32<<<(nbins + 255) / 256, 256, 0, stream>>>(counts, nbins);
    gabor_bin_atoms<<<(n_atoms + 255) / 256, 256, 0, stream>>>(
        tau, sg, n_atoms, ns, nbins, counts, lists);
    gabor_render_binned<<<nbins, BIN, 0, stream>>>(
        amp, tau, om, sg, ph, gm, counts, lists, out, ns);
  } else {
    gabor_zero_f32<<<(ns + 255) / 256, 256, 0, stream>>>(out, ns);
    gabor_render_atomic<<<n_atoms, 256, 0, stream>>>(
        amp, tau, om, sg, ph, gm, out, ns);
  }
}